// ModelNew_80908593923152
// MI455X (gfx1250) — compile-verified
//
#include <hip/hip_runtime.h>
#include <hip/hip_bf16.h>

// CDNA5 / gfx1250: fused ConvTranspose3d(s=2,p=2,k=5) + maxpool2 + maxpool3 + channel-sum
// Implicit GEMM on V_WMMA_F32_16X16X4_F32 with stride-parity tap decomposition.

typedef __attribute__((ext_vector_type(2))) float v2f;
typedef __attribute__((ext_vector_type(8))) float v8f;

#define N_BATCH 16
#define CIN     32
#define COUT    64
#define D_IN    16
#define H_IN    32
#define W_IN    32

// ---------------- repack kernels (one-shot, L2-resident afterwards) ----------------

// x (n,ci,d,h,w) -> x_t (n,d,h,w,ci)   channels-last: per-lane b64 A-fragment loads
__global__ __launch_bounds__(256) void repack_x_kernel(const float* __restrict__ x,
                                                       float* __restrict__ xt) {
    size_t i = (size_t)blockIdx.x * 256 + threadIdx.x;
    const size_t total = (size_t)N_BATCH * CIN * D_IN * H_IN * W_IN;
    if (i >= total) return;
    int w = (int)(i % W_IN); size_t r = i / W_IN;
    int h = (int)(r % H_IN); r /= H_IN;
    int d = (int)(r % D_IN); r /= D_IN;
    int ci = (int)(r % CIN);
    int n  = (int)(r / CIN);
    xt[((((size_t)n * D_IN + d) * H_IN + h) * W_IN + w) * CIN + ci] = x[i];
}

// w (ci,co,kd,kh,kw) -> w_t2 [tap][ci/2][co][2]  : per-lane b64 B-fragment loads
__global__ __launch_bounds__(256) void repack_w_kernel(const float* __restrict__ wsrc,
                                                       float* __restrict__ wt2) {
    int i = blockIdx.x * 256 + threadIdx.x;       // over 32*64*125
    if (i >= CIN * COUT * 125) return;
    int tap = i % 125; int r = i / 125;
    int co  = r % COUT;
    int ci  = r / COUT;
    wt2[(((size_t)tap * 16 + (ci >> 1)) * COUT + co) * 2 + (ci & 1)] = wsrc[i];
}

// ---------------- fused main kernel ----------------

__device__ inline void atomicMaxF32(float* addr, float val) {
    unsigned int* ua = (unsigned int*)addr;
    unsigned int cur = __float_as_uint(*addr);
    while (__uint_as_float(cur) < val) {
        unsigned int prev = atomicCAS(ua, cur, __float_as_uint(val));
        if (prev == cur) break;
        cur = prev;
    }
}

// grid: 800 = 16 n * 5 od' * 10 oh'; block: 256 (8 waves)
__global__ __launch_bounds__(256) void convt_pool_sum_kernel(
        const float* __restrict__ xt,    // [16][16][32][32][32ci]
        const float* __restrict__ wt2,   // [125][16][64][2]
        const float* __restrict__ bias,  // [64]
        float* __restrict__ out)         // [16][5][10][10]
{
    __shared__ float pool[10 * COUT];    // per-(ow', co) running max

    const int wg  = blockIdx.x;
    const int n   = wg / 50;
    const int rem = wg % 50;
    const int odp = rem / 10;            // od' 0..4
    const int ohp = rem % 10;            // oh' 0..9

    const int tid   = threadIdx.x;
    const int wave  = tid >> 5;          // 0..7 (wave32)
    const int lane  = tid & 31;
    const int lhalf = lane >> 4;         // A/B K-half:  lanes 0-15 -> K0/K1, 16-31 -> K2/K3
    const int l15   = lane & 15;         // A row index / B column index

    for (int i = tid; i < 10 * COUT; i += 256) pool[i] = -3.402823466e38f;
    __syncthreads();

    const v2f vzero = {0.0f, 0.0f};

    // 144 M-tiles: (dl 0..5) x (hl 0..5) x (w-parity 0..1) x (w-half 0..1)
    for (int t = wave; t < 144; t += 8) {
        const int wt = t & 1;
        const int wp = (t >> 1) & 1;
        const int hs = t >> 2;
        const int hl = hs % 6;
        const int dl = hs / 6;
        const int od = 6 * odp + dl;     // conv-output d, 0..29
        const int oh = 6 * ohp + hl;     // conv-output h, 0..59

        // this lane's A-row maps to conv-output w:
        const int wrow = wp + 2 * (16 * wt + l15);   // 0..63 (>=60 rows are padding)

        v8f acc0 = {}, acc1 = {}, acc2 = {}, acc3 = {};

        const int pd = od & 1, ph = oh & 1, pw = wp;
        for (int kd = pd; kd < 5; kd += 2) {
            const int id = (od + 2 - kd) >> 1;       // <= 15 always
            if (id < 0) continue;                    // uniform per tile
            for (int kh = ph; kh < 5; kh += 2) {
                const int ih = (oh + 2 - kh) >> 1;   // <= 30 always
                if (ih < 0) continue;                // uniform per tile
                const float* xbaseDH =
                    xt + ((((size_t)n * D_IN + id) * H_IN + ih) * W_IN) * CIN;
                for (int kw = pw; kw < 5; kw += 2) {
                    const int iw = (wrow + 2 - kw) >> 1;       // per-lane
                    const bool okw = ((unsigned)iw) < (unsigned)W_IN;
                    const int iwc = okw ? iw : 0;
                    const float* aPtr = xbaseDH + iwc * CIN + lhalf * 2;
                    const int tap = (kd * 5 + kh) * 5 + kw;
                    const float* wTap = wt2 + (size_t)tap * (16 * COUT * 2);

#pragma unroll
                    for (int kb = 0; kb < 8; ++kb) {           // K = ci, 4 per WMMA
                        v2f a = *(const v2f*)(aPtr + kb * 4);
                        a = okw ? a : vzero;                   // per-lane w-edge zeroing
                        const float* wk = wTap + ((kb * 2 + lhalf) * COUT + l15) * 2;
                        v2f b0 = *(const v2f*)(wk);            // co 0..15
                        v2f b1 = *(const v2f*)(wk + 32);       // co 16..31
                        v2f b2 = *(const v2f*)(wk + 64);       // co 32..47
                        v2f b3 = *(const v2f*)(wk + 96);       // co 48..63
                        acc0 = __builtin_amdgcn_wmma_f32_16x16x4_f32(
                            false, a, false, b0, (short)0, acc0, false, false);
                        acc1 = __builtin_amdgcn_wmma_f32_16x16x4_f32(
                            false, a, false, b1, (short)0, acc1, false, false);
                        acc2 = __builtin_amdgcn_wmma_f32_16x16x4_f32(
                            false, a, false, b2, (short)0, acc2, false, false);
                        acc3 = __builtin_amdgcn_wmma_f32_16x16x4_f32(
                            false, a, false, b3, (short)0, acc3, false, false);
                    }
                }
            }
        }

        // Fused pooling: D layout VGPR v -> row M = v + 8*lhalf, lane -> N = co.
        // Max-merge each conv value into pool[w/6][co]; skip pad rows (w >= 60).
#pragma unroll
        for (int v = 0; v < 8; ++v) {
            const int M = v + 8 * lhalf;
            const int w = wp + 2 * (16 * wt + M);
            if (w < 60) {
                const int base = (w / 6) * COUT + l15;
                atomicMaxF32(&pool[base +  0], acc0[v]);
                atomicMaxF32(&pool[base + 16], acc1[v]);
                atomicMaxF32(&pool[base + 32], acc2[v]);
                atomicMaxF32(&pool[base + 48], acc3[v]);
            }
        }
    }
    __syncthreads();

    // Channel sum (bias commutes with max: sum(max+b) = sum(max)+sum(b))
    if (tid < 10) {
        float s = 0.0f;
        for (int c = 0; c < COUT; ++c) s += pool[tid * COUT + c] + bias[c];
        out[((n * 5 + odp) * 10 + ohp) * 10 + tid] = s;
    }
}

// ---------------- launch ----------------

extern "C" void kernel_launch(void* const* d_in, const int* in_sizes, int n_in,
                              void* d_out, int out_size, void* d_ws, size_t ws_size,
                              hipStream_t stream) {
    const float* x    = (const float*)d_in[0];   // (16,32,16,32,32)
    const float* wsrc = (const float*)d_in[1];   // (32,64,5,5,5)
    const float* bias = (const float*)d_in[2];   // (64,)
    float* out = (float*)d_out;                  // (16,1,5,10,10) = 8000

    const size_t x_elems = (size_t)N_BATCH * CIN * D_IN * H_IN * W_IN;   // 8,388,608
    float* xt  = (float*)d_ws;
    float* wt2 = (float*)((char*)d_ws + x_elems * sizeof(float));        // +33.5 MB

    repack_x_kernel<<<(int)((x_elems + 255) / 256), 256, 0, stream>>>(x, xt);
    repack_w_kernel<<<(CIN * COUT * 125 + 255) / 256, 256, 0, stream>>>(wsrc, wt2);
    convt_pool_sum_kernel<<<16 * 5 * 10, 256, 0, stream>>>(xt, wt2, bias, out);
}